// TokenEmbedding_39608188404000
// MI455X (gfx1250) — compile-verified
//
#include <hip/hip_runtime.h>

// MI455X / gfx1250: wave32, WMMA. Full-fp32 path via V_WMMA_F32_16X16X4_F32,
// async global->LDS staging (ASYNCcnt) where the toolchain exposes it.

typedef float v2f __attribute__((ext_vector_type(2)));
typedef float v8f __attribute__((ext_vector_type(8)));

#define B_   48
#define W_   8192
#define F_   64
#define D_   128
#define BDIM 256
#define NWAVE 8

// ---- async global->LDS (gfx1250 GLOBAL_LOAD_ASYNC_TO_LDS_B128) ------------
// Probed signature: (int4 addrspace(1)*, int4 addrspace(3)*, imm, imm)
#if defined(__gfx1250__) && __has_builtin(__builtin_amdgcn_global_load_async_to_lds_b128)
typedef int i128_t __attribute__((vector_size(16)));
#define ASYNC_CP16(dst_lds, src_glb)                                           \
  __builtin_amdgcn_global_load_async_to_lds_b128(                              \
      (__attribute__((address_space(1))) i128_t*)(src_glb),                    \
      (__attribute__((address_space(3))) i128_t*)(dst_lds), 0, 0)
#if __has_builtin(__builtin_amdgcn_s_wait_asynccnt)
#define WAIT_ASYNC() __builtin_amdgcn_s_wait_asynccnt(0)
#else
#define WAIT_ASYNC() asm volatile("s_wait_asynccnt 0x0" ::: "memory")
#endif
#else
#define ASYNC_CP16(dst_lds, src_glb)                                           \
  (*(float4*)(dst_lds) = *(const float4*)(src_glb))
#define WAIT_ASYNC()
#endif

__device__ __forceinline__ v8f wmma_f32(v2f a, v2f b, v8f c) {
  // 8 args: (neg_a, A, neg_b, B, c_mod, C, reuse_a, reuse_b)
  return __builtin_amdgcn_wmma_f32_16x16x4_f32(false, a, false, b, (short)0, c,
                                               false, false);
}

// --------------------------------------------------------------------------
// Kernel 1: per-batch Gram matrix of row-normalized x.
//   gram[b,f,g] = sum_w (x[b,w,f]/|x[b,w,:]|) * (x[b,w,g]/|x[b,w,:]|)
// One block per batch. Double-buffered 64-row chunks: async-load chunk c+1
// into the alternate LDS buffer while chunk c is normalized and reduced with
// f32 WMMA. Each wave owns one fA row-tile and two fB tiles (A reused 2x).
// --------------------------------------------------------------------------
#define CH  64
#define NCH (W_ / CH)

__global__ void __launch_bounds__(BDIM)
gram_kernel(const float* __restrict__ x, float* __restrict__ gram) {
  __shared__ float xs[2][CH * F_];
  __shared__ float invn[CH];

  const int b    = blockIdx.x;
  const int tid  = threadIdx.x;
  const int wave = tid >> 5;
  const int lane = tid & 31;
  const int half = lane >> 4;   // 0: K=0..1, 1: K=2..3
  const int l16  = lane & 15;
  const int kb   = 2 * half;

  const int fA0 = (wave >> 1) * 16;        // shared A row-tile
  const int fB0 = (wave & 1) * 32;         // two B column-tiles: fB0, fB0+16

  const float* xb = x + (long)b * W_ * F_;

  // preload chunk 0
  for (int i = tid; i < CH * F_ / 4; i += BDIM)
    ASYNC_CP16(&xs[0][i * 4], xb + i * 4);
  WAIT_ASYNC();
  __syncthreads();

  v8f acc0 = {}, acc1 = {};

  for (int c = 0; c < NCH; ++c) {
    float* cur = xs[c & 1];

    if (c + 1 < NCH) {                     // async-fill the other buffer
      const float* xn  = xb + (long)(c + 1) * CH * F_;
      float*       nxt = xs[(c + 1) & 1];
      for (int i = tid; i < CH * F_ / 4; i += BDIM)
        ASYNC_CP16(&nxt[i * 4], xn + i * 4);
    }
    if (c + 2 < NCH && tid < CH * F_ / 32) // warm L2 for the chunk after
      __builtin_prefetch(xb + (long)(c + 2) * CH * F_ + tid * 32, 0, 0);

    if (tid < CH) {
      float s = 0.f;
#pragma unroll
      for (int f = 0; f < F_; ++f) {
        float v = cur[tid * F_ + f];
        s += v * v;
      }
      invn[tid] = rsqrtf(s);
    }
    __syncthreads();

    for (int i = tid; i < CH * F_ / 4; i += BDIM) {
      float  sc = invn[i >> 4];            // 16 float4 per row
      float4 v  = ((float4*)cur)[i];
      v.x *= sc; v.y *= sc; v.z *= sc; v.w *= sc;
      ((float4*)cur)[i] = v;
    }
    __syncthreads();

    for (int k0 = 0; k0 < CH; k0 += 4) {
      const int r0 = (k0 + kb) * F_;
      const int r1 = r0 + F_;
      v2f a, b0, b1;
      a.x  = cur[r0 + fA0 + l16];      a.y  = cur[r1 + fA0 + l16];
      b0.x = cur[r0 + fB0 + l16];      b0.y = cur[r1 + fB0 + l16];
      b1.x = cur[r0 + fB0 + 16 + l16]; b1.y = cur[r1 + fB0 + 16 + l16];
      acc0 = wmma_f32(a, b0, acc0);
      acc1 = wmma_f32(a, b1, acc1);
    }

    WAIT_ASYNC();                          // chunk c+1 landed in LDS
    __syncthreads();
  }

  float* g = gram + (long)b * F_ * F_;
#pragma unroll
  for (int j = 0; j < 8; ++j) {
    g[(fA0 + j + 8 * half) * F_ + fB0 + l16]      = acc0[j];
    g[(fA0 + j + 8 * half) * F_ + fB0 + 16 + l16] = acc1[j];
  }
}

// --------------------------------------------------------------------------
// Kernel 2 (fused): per 128-row output tile of one batch.
//   Phase A: async-stage x halo rows (148) into LDS union buffer.
//   Phase B: per wave-column, accumulate xse (K=64, gram) and xc (K=192,
//            conv-as-GEMM from LDS x) for 4-5 M-tiles with B-fragment reuse;
//            fold h = xc + ctx_b + relu(xse + seW_b) into registers.
//   Phase C: barrier; write h fragments into the same LDS buffer.
//   Phase D: tokenConv out = H3(128x192) x TW(192x128); each wave owns one
//            16-wide d column over all 8 M-tiles (B reused 8x per K-step).
// --------------------------------------------------------------------------
#define TILE   128
#define XSLOTS 148   // x rows w0-2 .. w0+145 (circular)
#define HSLOTS 144   // 9 M-tiles of h rows (130 live + pad)

template <int NM, int MBASE>
__device__ __forceinline__ void stage1_col(
    float* __restrict__ buf, const float* __restrict__ seW_w,
    const float* __restrict__ gb, const float* __restrict__ ctx_w,
    const float* __restrict__ ctx_b, const float* __restrict__ seW_b,
    int w0, int f0, int l16, int half, int kb) {
  v8f accS[NM], accC[NM];
#pragma unroll
  for (int mi = 0; mi < NM; ++mi) { accS[mi] = {}; accC[mi] = {}; }

  int hw[NM];
#pragma unroll
  for (int mi = 0; mi < NM; ++mi)
    hw[mi] = (w0 - 1 + (MBASE + mi) * 16 + l16) & (W_ - 1);

  // ---- xse: K = 64 over g; B = gram[f0+l16, g] reused across M-tiles ----
  const float* bpg = gb + (f0 + l16) * F_ + kb;
  for (int g0 = 0; g0 < F_; g0 += 4) {
    v2f bf;
    bf.x = bpg[g0]; bf.y = bpg[g0 + 1];
#pragma unroll
    for (int mi = 0; mi < NM; ++mi) {
      const float* ap = seW_w + (long)hw[mi] * F_ + g0 + kb;
      v2f a; a.x = ap[0]; a.y = ap[1];
      accS[mi] = wmma_f32(a, bf, accS[mi]);
    }
  }

  // ---- xc: K = 192 (3 taps x 64 ch); A from LDS x tile ----
#pragma unroll
  for (int tap = 0; tap < 3; ++tap) {
    const float* brow = ctx_w + (f0 + l16) * (F_ * 3) + kb * 3 + tap;
    for (int g0 = 0; g0 < F_; g0 += 4) {
      v2f bf;
      bf.x = brow[g0 * 3]; bf.y = brow[g0 * 3 + 3];
#pragma unroll
      for (int mi = 0; mi < NM; ++mi) {
        const float* ap =
            buf + ((MBASE + mi) * 16 + l16 + tap) * F_ + g0 + kb;
        v2f a; a.x = ap[0]; a.y = ap[1];
        accC[mi] = wmma_f32(a, bf, accC[mi]);
      }
    }
  }

  // ---- fold h = xc + ctx_b + relu(xse + seW_b) into registers ----
  const float cb = ctx_b[f0 + l16];
#pragma unroll
  for (int mi = 0; mi < NM; ++mi) {
#pragma unroll
    for (int j = 0; j < 8; ++j) {
      const int   slot = (MBASE + mi) * 16 + j + 8 * half;
      const int   hwj  = (w0 - 1 + slot) & (W_ - 1);
      float       se   = accS[mi][j] + seW_b[hwj];
      se = se > 0.f ? se : 0.f;
      accS[mi][j] = accC[mi][j] + cb + se;
    }
  }

  // all waves must be done READING the x tile before h overwrites it
  __syncthreads();

#pragma unroll
  for (int mi = 0; mi < NM; ++mi) {
#pragma unroll
    for (int j = 0; j < 8; ++j) {
      const int slot = (MBASE + mi) * 16 + j + 8 * half;
      buf[slot * F_ + f0 + l16] = accS[mi][j];
    }
  }
}

__global__ void __launch_bounds__(BDIM)
fused_kernel(const float* __restrict__ x,
             const float* __restrict__ ctx_w, const float* __restrict__ ctx_b,
             const float* __restrict__ token_w,
             const float* __restrict__ seW_w, const float* __restrict__ seW_b,
             const float* __restrict__ gram, float* __restrict__ out) {
  __shared__ float buf[XSLOTS * F_];   // x halo tile, then h tile (union)

  const int b    = blockIdx.x >> 6;    // 64 tiles per batch
  const int tile = blockIdx.x & 63;
  const int w0   = tile * TILE;
  const int tid  = threadIdx.x;
  const int wave = tid >> 5;
  const int lane = tid & 31;
  const int half = lane >> 4;
  const int l16  = lane & 15;
  const int kb   = 2 * half;

  const float* xb = x + (long)b * W_ * F_;
  const float* gb = gram + (long)b * F_ * F_;

  // ---- Phase A: async stage of x rows (w0-2 .. w0+145, circular) ----
  for (int i = tid; i < XSLOTS * F_ / 4; i += BDIM) {
    const int s  = i >> 4;                    // 16 float4 per row
    const int xr = (w0 - 2 + s) & (W_ - 1);
    ASYNC_CP16(&buf[i * 4], xb + (long)xr * F_ + (i & 15) * 4);
  }
  WAIT_ASYNC();
  __syncthreads();

  // ---- Phases B+C: waves 0-3 -> M-tiles 0..4, waves 4-7 -> M-tiles 5..8;
  //      each wave owns one 16-wide f column ----
  const int f0 = (wave & 3) * 16;
  if (wave < 4)
    stage1_col<5, 0>(buf, seW_w, gb, ctx_w, ctx_b, seW_b, w0, f0, l16, half, kb);
  else
    stage1_col<4, 5>(buf, seW_w, gb, ctx_w, ctx_b, seW_b, w0, f0, l16, half, kb);
  __syncthreads();

  // ---- Phase D: tokenConv; wave = one d column over all 8 M-tiles ----
  const int d0 = wave * 16;
  v8f acc[8];
#pragma unroll
  for (int mi = 0; mi < 8; ++mi) acc[mi] = {};

#pragma unroll
  for (int tap = 0; tap < 3; ++tap) {
    const float* brow = token_w + (d0 + l16) * (F_ * 3) + kb * 3 + tap;
    for (int g0 = 0; g0 < F_; g0 += 4) {
      v2f bf;
      bf.x = brow[g0 * 3]; bf.y = brow[g0 * 3 + 3];
#pragma unroll
      for (int mi = 0; mi < 8; ++mi) {
        // out row r = mi*16+m reads h slot r+tap (buf slot s <-> hw w0-1+s)
        const float* ap = buf + (mi * 16 + l16 + tap) * F_ + g0 + kb;
        v2f a; a.x = ap[0]; a.y = ap[1];
        acc[mi] = wmma_f32(a, bf, acc[mi]);
      }
    }
  }

#pragma unroll
  for (int mi = 0; mi < 8; ++mi) {
    float* orow = out + ((long)b * W_ + w0 + mi * 16) * D_ + d0;
#pragma unroll
    for (int j = 0; j < 8; ++j)
      orow[(j + 8 * half) * D_ + l16] = acc[mi][j];
  }
}

// --------------------------------------------------------------------------
extern "C" void kernel_launch(void* const* d_in, const int* in_sizes, int n_in,
                              void* d_out, int out_size, void* d_ws, size_t ws_size,
                              hipStream_t stream) {
  const float* x       = (const float*)d_in[0];
  const float* ctx_w   = (const float*)d_in[1];
  const float* ctx_b   = (const float*)d_in[2];
  const float* token_w = (const float*)d_in[3];
  const float* seW_w   = (const float*)d_in[4];
  const float* seW_b   = (const float*)d_in[5];
  float* out  = (float*)d_out;
  float* gram = (float*)d_ws;   // B*64*64 floats = 768 KiB

  hipLaunchKernelGGL(gram_kernel, dim3(B_), dim3(BDIM), 0, stream, x, gram);
  hipLaunchKernelGGL(fused_kernel, dim3(B_ * (W_ / TILE)), dim3(BDIM), 0, stream,
                     x, ctx_w, ctx_b, token_w, seW_w, seW_b, gram, out);
}